// adder2d_1580547972590
// MI455X (gfx1250) — compile-verified
//
#include <hip/hip_runtime.h>

// Adder2D (AdderNet) forward value: out[b,f,h,w] = -sum_k |W[f,k] - patch[b,k,(h,w)]|
// B=16, C=128, H=W=28, F=128, 3x3 conv geometry, stride 1, pad 1.
//
// WMMA is mathematically inapplicable (L1 distance is not bilinear), so this
// kernel targets MI455X via: f32 VALU core using free |src| modifiers
// (v_add_f32 acc,acc,|t|), CDNA5 Tensor Data Mover (tensor_load_to_lds) for
// strided weight staging, s_wait_tensorcnt + barriers for double buffering.

#define BB 16
#define CC 128
#define HH 28
#define WW 28
#define FF 128
#define KK 1152          // C*3*3
#define RROWS 2          // output rows per block
#define CCH 4            // channels staged per chunk
#define NCHUNK (CC / CCH)                     // 32
#define XS_ELEMS (CCH * (RROWS + 2) * WW)     // 4*4*28 = 448
#define WROW (CCH * 9)                        // 36 floats per filter per chunk
#define WS_ELEMS (FF * WROW)                  // 128*36 = 4608

typedef unsigned int v4u __attribute__((ext_vector_type(4)));
typedef int v8i __attribute__((ext_vector_type(8)));
typedef int v4i __attribute__((ext_vector_type(4)));

// One TDM 2D tile load: rows = 128 filters, row length = CCH*9 = 36 f32
// elements starting at W + c0*9, row stride = 1152 elements. Lands compact
// [128][36] at lds_off.  Descriptor per CDNA5 ISA section 8.3/8.4.
__device__ __forceinline__ void tdm_load_W_chunk(const float* Wg, int c0,
                                                 unsigned lds_off) {
  unsigned long long ga =
      (unsigned long long)(size_t)(const void*)(Wg + (size_t)c0 * 9);
  v4u g0;
  g0[0] = 1u;                                         // count=1, user mode
  g0[1] = lds_off;                                    // lds_addr (bytes)
  g0[2] = (unsigned)(ga & 0xffffffffull);             // global_addr[31:0]
  g0[3] = (unsigned)((ga >> 32) & 0x01ffffffull)      // global_addr[56:32]
          | (2u << 30);                               // type = 2 ("image")
  v8i g1;
  g1[0] = (int)(2u << 16);                 // data_size = 2 -> 4 bytes
  g1[1] = (int)((unsigned)WROW << 16);     // tensor_dim0[15:0] in [31:16]
  g1[2] = (int)((unsigned)FF << 16);       // tensor_dim0 hi=0 | tensor_dim1 lo
  g1[3] = (int)((unsigned)WROW << 16);     // tensor_dim1 hi=0 | tile_dim0=36
  g1[4] = FF;                              // tile_dim1=128, tile_dim2=0
  g1[5] = KK;                              // tensor_dim0_stride low32 = 1152
  g1[6] = 0;                               // stride0 hi | stride1 lo
  g1[7] = 0;                               // stride1 hi
  v4i g2 = {0, 0, 0, 0};                   // 2D tile: groups 2/3 unused
  v4i g3 = {0, 0, 0, 0};
#if defined(__clang_major__) && (__clang_major__ >= 23)
  v8i g4 = {0, 0, 0, 0, 0, 0, 0, 0};
  __builtin_amdgcn_tensor_load_to_lds(g0, g1, g2, g3, g4, 0);
#else
  __builtin_amdgcn_tensor_load_to_lds(g0, g1, g2, g3, 0);
#endif
}

__global__ __launch_bounds__(256, 1) void adder2d_l1_kernel(
    const float* __restrict__ x, const float* __restrict__ W,
    float* __restrict__ out) {
  __shared__ float ws[2][WS_ELEMS];   // 36,864 B  (W chunk, double buffered)
  __shared__ float xs[2][XS_ELEMS];   //  3,584 B  (x slab,  double buffered)

  const int tid  = threadIdx.x;
  const int lane = tid & 31;
  const int wid  = tid >> 5;
  const int b    = blockIdx.y;
  const int r0   = blockIdx.x * RROWS;        // first output row of this block
  const int f0   = wid * 16;                  // 16 filters per wave
  const int col  = (lane < WW) ? lane : (WW - 1);   // lanes 28..31: clamped, masked at store
  const int colm = (col > 0) ? (col - 1) : 0;
  const int colp = (col < WW - 1) ? (col + 1) : (WW - 1);
  const bool leftok  = (col > 0);
  const bool rightok = (col < WW - 1);
  // Force a *scalar* branch so only wave 0 issues the TDM (TDM ignores EXEC).
  const bool leader = (__builtin_amdgcn_readfirstlane(wid) == 0);

  const size_t xbase = (size_t)b * CC * HH * WW;

  float acc[RROWS][16];
#pragma unroll
  for (int r = 0; r < RROWS; ++r)
#pragma unroll
    for (int f = 0; f < 16; ++f) acc[r][f] = 0.0f;

  // ---------------- prologue: stage chunk 0 into buffer 0 ----------------
  {
    float xv[2];
#pragma unroll
    for (int j = 0; j < 2; ++j) {
      int i = tid + j * 256;
      float v = 0.0f;
      if (i < XS_ELEMS) {
        int cl   = i / ((RROWS + 2) * WW);
        int rem  = i - cl * ((RROWS + 2) * WW);
        int rowl = rem / WW;
        int ccol = rem - rowl * WW;
        int h    = r0 - 1 + rowl;                    // zero padding rows
        if (h >= 0 && h < HH)
          v = x[xbase + (size_t)cl * (HH * WW) + h * WW + ccol];
      }
      xv[j] = v;
    }
    if (leader) tdm_load_W_chunk(W, 0, (unsigned)(size_t)(void*)&ws[0][0]);
#pragma unroll
    for (int j = 0; j < 2; ++j) {
      int i = tid + j * 256;
      if (i < XS_ELEMS) xs[0][i] = xv[j];
    }
    if (leader) __builtin_amdgcn_s_wait_tensorcnt(0);
    __syncthreads();
  }

  // ---------------- main K loop: 32 chunks of 4 channels ----------------
  for (int ci = 0; ci < NCHUNK; ++ci) {
    const int buf   = ci & 1;
    const bool more = (ci + 1) < NCHUNK;

    // Issue next chunk's global loads + TDM *before* compute (latency hidden).
    float xv[2];
    if (more) {
      const int c0n = (ci + 1) * CCH;
#pragma unroll
      for (int j = 0; j < 2; ++j) {
        int i = tid + j * 256;
        float v = 0.0f;
        if (i < XS_ELEMS) {
          int cl   = i / ((RROWS + 2) * WW);
          int rem  = i - cl * ((RROWS + 2) * WW);
          int rowl = rem / WW;
          int ccol = rem - rowl * WW;
          int h    = r0 - 1 + rowl;
          if (h >= 0 && h < HH)
            v = x[xbase + (size_t)(c0n + cl) * (HH * WW) + h * WW + ccol];
        }
        xv[j] = v;
      }
      if (leader)
        tdm_load_W_chunk(W, (ci + 1) * CCH,
                         (unsigned)(size_t)(void*)&ws[buf ^ 1][0]);
    }

    // ---- compute current chunk from LDS ----
    const float* xsb = &xs[buf][0];
    const float* wsb = &ws[buf][0];
    for (int cl = 0; cl < CCH; ++cl) {
      const float* xc = xsb + cl * (RROWS + 2) * WW;
      float pA[RROWS + 2][3];
#pragma unroll
      for (int rr = 0; rr < RROWS + 2; ++rr) {
        float lv = xc[rr * WW + colm];
        float cv = xc[rr * WW + col];
        float rv = xc[rr * WW + colp];
        pA[rr][0] = leftok ? lv : 0.0f;    // zero padding, left column
        pA[rr][1] = cv;
        pA[rr][2] = rightok ? rv : 0.0f;   // zero padding, right column
      }
      const float* wb = wsb + (size_t)f0 * WROW + cl * 9;
#pragma unroll
      for (int fl = 0; fl < 16; ++fl) {
        const float* wf = wb + fl * WROW;  // wave-uniform -> LDS broadcast
        float w[9];
#pragma unroll
        for (int k = 0; k < 9; ++k) w[k] = wf[k];
#pragma unroll
        for (int r = 0; r < RROWS; ++r) {
          float s = acc[r][fl];
#pragma unroll
          for (int kh = 0; kh < 3; ++kh)
#pragma unroll
            for (int kw = 0; kw < 3; ++kw)
              s += __builtin_fabsf(w[kh * 3 + kw] - pA[r + kh][kw]);
          acc[r][fl] = s;
        }
      }
    }

    // ---- publish next buffers ----
    if (more) {
#pragma unroll
      for (int j = 0; j < 2; ++j) {
        int i = tid + j * 256;
        if (i < XS_ELEMS) xs[buf ^ 1][i] = xv[j];
      }
      if (leader) __builtin_amdgcn_s_wait_tensorcnt(0);
      __syncthreads();
    }
  }

  // ---------------- store: out = -L1 ----------------
  if (lane < WW) {
#pragma unroll
    for (int r = 0; r < RROWS; ++r) {
      const int h = r0 + r;
#pragma unroll
      for (int fl = 0; fl < 16; ++fl) {
        const size_t o = (((size_t)b * FF + (f0 + fl)) * HH + h) * WW + col;
        out[o] = -acc[r][fl];
      }
    }
  }
}

extern "C" void kernel_launch(void* const* d_in, const int* in_sizes, int n_in,
                              void* d_out, int out_size, void* d_ws,
                              size_t ws_size, hipStream_t stream) {
  (void)in_sizes; (void)n_in; (void)d_ws; (void)ws_size; (void)out_size;
  const float* x = (const float*)d_in[0];   // [16,128,28,28] f32
  const float* W = (const float*)d_in[1];   // [128,128,3,3]  f32
  float* out = (float*)d_out;               // [16,128,28,28] f32
  dim3 grid(HH / RROWS, BB);                // (14, 16) blocks
  adder2d_l1_kernel<<<grid, 256, 0, stream>>>(x, W, out);
}